// WalletGATN_82351702933634
// MI455X (gfx1250) — compile-verified
//
#include <hip/hip_runtime.h>
#include <math.h>

#define HIDDEN   128
#define HEADS    4
#define CDIM     32
#define LAYERS   3
#define NODE_DIM 12
#define EHID     64    // HIDDEN/2
#define BGRAPH   64

typedef __attribute__((ext_vector_type(2))) float v2f;
typedef __attribute__((ext_vector_type(8))) float v8f;

__device__ __forceinline__ float gelu_exact(float x) {
  return 0.5f * x * (1.0f + erff(x * 0.70710678118654752440f));
}
// order-preserving float<->uint encoding for atomic max on signed floats
__device__ __forceinline__ unsigned fenc(float f) {
  unsigned u = __float_as_uint(f);
  return (u >> 31) ? ~u : (u | 0x80000000u);
}
__device__ __forceinline__ float fdec(unsigned u) {
  return __uint_as_float((u >> 31) ? (u & 0x7FFFFFFFu) : ~u);
}
#define ENC_NEG_INF 0x007FFFFFu  // fenc(-inf)

__device__ __forceinline__ void atomAddF(float* p, float v) {
  __hip_atomic_fetch_add(p, v, __ATOMIC_RELAXED, __HIP_MEMORY_SCOPE_AGENT);
}

__global__ void fill_u32(unsigned* __restrict__ p, unsigned val, long long n) {
  long long t = (long long)blockIdx.x * blockDim.x + threadIdx.x;
  if (t < n) p[t] = val;
}

// ---------------------------------------------------------------------------
// C[M,128] = A[M,K] @ B[K,128] (+bias). M%16==0, K%4==0. One wave per 16x16
// tile; 8 waves/block cover all 8 column tiles; grid.x = M/16.
// Uses native f32 WMMA (V_WMMA_F32_16X16X4_F32) -> exact f32 like reference.
// ---------------------------------------------------------------------------
__global__ void gemm_wmma_f32_n128(const float* __restrict__ A,
                                   const float* __restrict__ Bm,
                                   const float* __restrict__ bias,
                                   float* __restrict__ C, int M, int K) {
  const int Nn = 128;
  int lane = threadIdx.x & 31;
  int wv   = threadIdx.x >> 5;        // 0..7 : column tile
  int row0 = blockIdx.x * 16;
  int col0 = wv * 16;
  int r  = lane & 15;
  int kk = (lane >> 4) << 1;          // lanes 0-15 -> K=0,1 ; lanes 16-31 -> K=2,3
  v8f acc = {};
  for (int k = 0; k < K; k += 4) {
    v2f a, b;
    a[0] = A[(long long)(row0 + r) * K + (k + kk)];
    a[1] = A[(long long)(row0 + r) * K + (k + kk + 1)];
    b[0] = Bm[(long long)(k + kk) * Nn + col0 + r];
    b[1] = Bm[(long long)(k + kk + 1) * Nn + col0 + r];
    acc = __builtin_amdgcn_wmma_f32_16x16x4_f32(false, a, false, b, (short)0,
                                                acc, false, false);
  }
  // C/D layout: VGPR i -> M = i + (lane<16 ? 0 : 8), N = lane%16
  int rowOff = (lane >> 4) * 8;
  float bb = bias ? bias[col0 + r] : 0.0f;
  for (int i = 0; i < 8; ++i)
    C[(long long)(row0 + rowOff + i) * Nn + col0 + r] = acc[i] + bb;
}

// ---------------------------------------------------------------------------
// Row LayerNorm over 128 channels, wave32: 4 channels/lane, shfl reductions.
// out = LN(X (+R) (+biasPerChan)) * g + b ; optional exact GELU.
// ---------------------------------------------------------------------------
__global__ void ln_kernel(const float* __restrict__ X, const float* __restrict__ R,
                          const float* __restrict__ bias, const float* __restrict__ g,
                          const float* __restrict__ bvec, float* __restrict__ out,
                          int Nn, int do_gelu) {
  int lane = threadIdx.x & 31;
  int node = blockIdx.x * (blockDim.x >> 5) + (threadIdx.x >> 5);
  if (node >= Nn) return;
  long long base = (long long)node * 128;
  float v[4], s = 0.f;
  for (int j = 0; j < 4; ++j) {
    int c = j * 32 + lane;
    float x = X[base + c];
    if (R)    x += R[base + c];
    if (bias) x += bias[c];
    v[j] = x; s += x;
  }
  for (int o = 16; o > 0; o >>= 1) s += __shfl_xor(s, o, 32);
  float mean = s * (1.0f / 128.0f);
  float vs = 0.f;
  for (int j = 0; j < 4; ++j) { float d = v[j] - mean; vs += d * d; }
  for (int o = 16; o > 0; o >>= 1) vs += __shfl_xor(vs, o, 32);
  float inv = rsqrtf(vs * (1.0f / 128.0f) + 1e-5f);
  for (int j = 0; j < 4; ++j) {
    int c = j * 32 + lane;
    float y = (v[j] - mean) * inv * g[c] + bvec[c];
    if (do_gelu) y = gelu_exact(y);
    out[base + c] = y;
  }
}

// V[l][k][hd] = sum_c gat_edge_w[l][k][hd*32+c] * att_edge[l][hd][c]  (64x4 per layer)
__global__ void vmat_kernel(const float* __restrict__ gat_edge_w,
                            const float* __restrict__ att_edge,
                            float* __restrict__ V) {
  int t = blockIdx.x * blockDim.x + threadIdx.x;
  if (t >= LAYERS * EHID * HEADS) return;
  int l = t / (EHID * HEADS);
  int rem = t % (EHID * HEADS);
  int k = rem / HEADS, hd = rem % HEADS;
  const float* w  = gat_edge_w + ((long long)l * EHID + k) * HIDDEN + hd * CDIM;
  const float* ae = att_edge + (l * HEADS + hd) * CDIM;
  float s = 0.f;
  for (int c = 0; c < CDIM; ++c) s += w[c] * ae[c];
  V[t] = s;
}

// Per real edge: ea = GELU(edge_attr@ee_w + ee_b) (64 regs); accumulate
// self-loop mean numerator + degree; contract with V -> a_e[e][12].
__global__ void edge_kernel(const float* __restrict__ edge_attr,
                            const int* __restrict__ dstp,
                            const float* __restrict__ ee_w, const float* __restrict__ ee_b,
                            const float* __restrict__ V,
                            float* __restrict__ a_e, float* __restrict__ loop_sum,
                            float* __restrict__ deg, long long E) {
  long long e = (long long)blockIdx.x * blockDim.x + threadIdx.x;
  if (e >= E) return;
  float x0 = edge_attr[e * 4 + 0], x1 = edge_attr[e * 4 + 1];
  float x2 = edge_attr[e * 4 + 2], x3 = edge_attr[e * 4 + 3];
  int d = dstp[e];
  float t[EHID];
  for (int j = 0; j < EHID; ++j) {
    float s = x0 * ee_w[j] + x1 * ee_w[EHID + j] + x2 * ee_w[2 * EHID + j] +
              x3 * ee_w[3 * EHID + j] + ee_b[j];
    t[j] = gelu_exact(s);
  }
  atomAddF(&deg[d], 1.0f);
  for (int j = 0; j < EHID; ++j) atomAddF(&loop_sum[(long long)d * EHID + j], t[j]);
  for (int lh = 0; lh < LAYERS * HEADS; ++lh) {
    int l = lh >> 2, hd = lh & 3;
    float s = 0.f;
    for (int j = 0; j < EHID; ++j) s += t[j] * V[(l * EHID + j) * HEADS + hd];
    a_e[e * 12 + lh] = s;
  }
}

// Self-loop rows: loop_attr = loop_sum / max(deg,1); contract with V.
__global__ void loop_kernel(const float* __restrict__ loop_sum,
                            const float* __restrict__ deg,
                            const float* __restrict__ V,
                            float* __restrict__ a_e, long long E, int Nn) {
  int n = blockIdx.x * blockDim.x + threadIdx.x;
  if (n >= Nn) return;
  float invd = 1.0f / fmaxf(deg[n], 1.0f);
  float t[EHID];
  for (int j = 0; j < EHID; ++j) t[j] = loop_sum[(long long)n * EHID + j] * invd;
  for (int lh = 0; lh < LAYERS * HEADS; ++lh) {
    int l = lh >> 2, hd = lh & 3;
    float s = 0.f;
    for (int j = 0; j < EHID; ++j) s += t[j] * V[(l * EHID + j) * HEADS + hd];
    a_e[(E + n) * 12 + lh] = s;
  }
}

// a_s[n,h] = hh[n,h,:]·att_src[h] ; a_d likewise. One thread per (node,head).
__global__ void attn_scalar_kernel(const float* __restrict__ hh,
                                   const float* __restrict__ asv,
                                   const float* __restrict__ adv,
                                   float* __restrict__ a_s, float* __restrict__ a_d,
                                   int Nn) {
  long long t = (long long)blockIdx.x * blockDim.x + threadIdx.x;
  if (t >= (long long)Nn * HEADS) return;
  int n = (int)(t >> 2), hd = (int)(t & 3);
  const float* row = hh + (long long)n * HIDDEN + hd * CDIM;
  float s = 0.f, d = 0.f;
  for (int c = 0; c < CDIM; ++c) {
    float v = row[c];
    s += v * asv[hd * CDIM + c];
    d += v * adv[hd * CDIM + c];
  }
  a_s[t] = s; a_d[t] = d;
}

// pass 1: alpha = leaky_relu(a_s[s]+a_d[d]+a_e); segment-max via uint atomics
__global__ void alpha1_kernel(const int* __restrict__ src, const int* __restrict__ dstp,
                              const float* __restrict__ a_s, const float* __restrict__ a_d,
                              const float* __restrict__ a_e, int layer,
                              float* __restrict__ alpha, unsigned* __restrict__ mEnc,
                              long long E, int Nn) {
  long long t = (long long)blockIdx.x * blockDim.x + threadIdx.x;
  if (t >= (E + Nn) * 4) return;
  long long i = t >> 2; int hd = (int)(t & 3);
  int s, d;
  if (i < E) { s = src[i]; d = dstp[i]; } else { s = d = (int)(i - E); }
  float al = a_s[(long long)s * 4 + hd] + a_d[(long long)d * 4 + hd] +
             a_e[i * 12 + layer * 4 + hd];
  al = al > 0.f ? al : 0.2f * al;
  alpha[t] = al;
  atomicMax(&mEnc[(long long)d * 4 + hd], fenc(al));
}

// pass 2: ex = exp(alpha - m[d]); segment-sum denom
__global__ void alpha2_kernel(const int* __restrict__ dstp,
                              float* __restrict__ alpha,
                              const unsigned* __restrict__ mEnc,
                              float* __restrict__ denom, long long E, int Nn) {
  long long t = (long long)blockIdx.x * blockDim.x + threadIdx.x;
  if (t >= (E + Nn) * 4) return;
  long long i = t >> 2; int hd = (int)(t & 3);
  int d = (i < E) ? dstp[i] : (int)(i - E);
  float ex = expf(alpha[t] - fdec(mEnc[(long long)d * 4 + hd]));
  alpha[t] = ex;
  atomAddF(&denom[(long long)d * 4 + hd], ex);
}

// pass 3: agg[d,h,:] += (ex/denom) * hh[s,h,:]   (32 f32 atomics / (edge,head))
__global__ void agg_kernel(const int* __restrict__ src, const int* __restrict__ dstp,
                           const float* __restrict__ alpha, const float* __restrict__ denom,
                           const float* __restrict__ hh, float* __restrict__ agg,
                           long long E, int Nn) {
  long long t = (long long)blockIdx.x * blockDim.x + threadIdx.x;
  if (t >= (E + Nn) * 4) return;
  long long i = t >> 2; int hd = (int)(t & 3);
  int s, d;
  if (i < E) { s = src[i]; d = dstp[i]; } else { s = d = (int)(i - E); }
  float w = alpha[t] / (denom[(long long)d * 4 + hd] + 1e-16f);
  const float* hr = hh + (long long)s * HIDDEN + hd * CDIM;
  float* ar = agg + (long long)d * HIDDEN + hd * CDIM;
  for (int c = 0; c < CDIM; ++c) atomAddF(&ar[c], w * hr[c]);
}

__global__ void pool_acc_kernel(const float* __restrict__ h, const int* __restrict__ batch,
                                float* __restrict__ mean_sum, unsigned* __restrict__ maxEnc,
                                float* __restrict__ cnt, int Nn) {
  long long t = (long long)blockIdx.x * blockDim.x + threadIdx.x;
  if (t >= (long long)Nn * 128) return;
  int n = (int)(t >> 7), c = (int)(t & 127);
  int b = batch[n];
  float v = h[t];
  atomAddF(&mean_sum[b * 128 + c], v);
  atomicMax(&maxEnc[b * 128 + c], fenc(v));
  if (c == 0) atomAddF(&cnt[b], 1.0f);
}

__global__ void pool_final_kernel(const float* __restrict__ mean_sum,
                                  const unsigned* __restrict__ maxEnc,
                                  const float* __restrict__ cnt, float* __restrict__ g) {
  int t = blockIdx.x * blockDim.x + threadIdx.x;
  if (t >= BGRAPH * 256) return;
  int b = t >> 8, c = t & 255;
  float cn = cnt[b];
  float v;
  if (c < 128) v = mean_sum[b * 128 + c] / fmaxf(cn, 1.0f);
  else         v = cn > 0.f ? fdec(maxEnc[b * 128 + c - 128]) : 0.f;
  g[t] = v;
}

// tiny dense: out[M,Nout] = act(in @ W + b); act 0=none,1=gelu,2=sigmoid
__global__ void dense_kernel(const float* __restrict__ in, const float* __restrict__ W,
                             const float* __restrict__ bvec, float* __restrict__ out,
                             int M, int Kin, int Nout, int act) {
  int t = blockIdx.x * blockDim.x + threadIdx.x;
  if (t >= M * Nout) return;
  int i = t / Nout, j = t % Nout;
  float s = bvec ? bvec[j] : 0.f;
  for (int k = 0; k < Kin; ++k) s += in[i * Kin + k] * W[k * Nout + j];
  if (act == 1)      s = gelu_exact(s);
  else if (act == 2) s = 1.0f / (1.0f + expf(-s));
  out[t] = s;
}

// ---------------------------------------------------------------------------
extern "C" void kernel_launch(void* const* d_in, const int* in_sizes, int n_in,
                              void* d_out, int out_size, void* d_ws, size_t ws_size,
                              hipStream_t stream) {
  const float* x          = (const float*)d_in[0];
  const float* edge_attr  = (const float*)d_in[1];
  const int*   edge_index = (const int*)d_in[2];
  const int*   batch      = (const int*)d_in[3];
  const float* enc_w      = (const float*)d_in[4];
  const float* enc_b      = (const float*)d_in[5];
  const float* enc_g      = (const float*)d_in[6];
  const float* enc_beta   = (const float*)d_in[7];
  const float* ee_w       = (const float*)d_in[8];
  const float* ee_b       = (const float*)d_in[9];
  const float* gat_lin_w  = (const float*)d_in[10];
  const float* gat_edge_w = (const float*)d_in[11];
  const float* att_src    = (const float*)d_in[12];
  const float* att_dst    = (const float*)d_in[13];
  const float* att_edge   = (const float*)d_in[14];
  const float* gat_bias   = (const float*)d_in[15];
  const float* norm_g     = (const float*)d_in[16];
  const float* norm_bv    = (const float*)d_in[17];
  const float* freq_w1    = (const float*)d_in[18];
  const float* freq_b1    = (const float*)d_in[19];
  const float* freq_w2    = (const float*)d_in[20];
  const float* freq_b2    = (const float*)d_in[21];
  const float* cls_w1     = (const float*)d_in[22];
  const float* cls_b1     = (const float*)d_in[23];
  const float* cls_w2     = (const float*)d_in[24];
  const float* cls_b2     = (const float*)d_in[25];

  const long long N = in_sizes[0] / NODE_DIM;   // 50000 (divisible by 16)
  const long long E = in_sizes[2] / 2;          // 1.6M
  const int* srcp = edge_index;
  const int* dstp = edge_index + E;

  // bump-allocate workspace (~197 MB total)
  char* wp = (char*)d_ws;
  auto alloc = [&](size_t bytes) -> char* {
    char* r = wp; wp += (bytes + 255) & ~(size_t)255; return r;
  };
  float*    h        = (float*)alloc(N * 128 * 4);
  float*    hh       = (float*)alloc(N * 128 * 4);
  float*    agg      = (float*)alloc(N * 128 * 4);
  float*    a_e      = (float*)alloc((E + N) * 12 * 4);
  float*    alpha    = (float*)alloc((E + N) * 4 * 4);
  float*    a_s      = (float*)alloc(N * 4 * 4);
  float*    a_d      = (float*)alloc(N * 4 * 4);
  unsigned* mEnc     = (unsigned*)alloc(N * 4 * 4);
  float*    denom    = (float*)alloc(N * 4 * 4);
  float*    deg      = (float*)alloc(N * 4);
  float*    loop_sum = (float*)alloc(N * EHID * 4);
  float*    V        = (float*)alloc(LAYERS * EHID * HEADS * 4);
  float*    mean_sum = (float*)alloc(BGRAPH * 128 * 4);
  unsigned* maxEnc   = (unsigned*)alloc(BGRAPH * 128 * 4);
  float*    cnt      = (float*)alloc(BGRAPH * 4);
  float*    gbuf     = (float*)alloc(BGRAPH * 256 * 4);
  float*    t1       = (float*)alloc(BGRAPH * 128 * 4);
  float*    t2       = (float*)alloc(BGRAPH * 64 * 4);
  float*    t3       = (float*)alloc(BGRAPH * 64 * 4);
  (void)ws_size; (void)n_in; (void)out_size;

  auto nb = [](long long n, int b) { return (unsigned)((n + b - 1) / b); };

  // edge features -> self-loop attrs + pre-contracted attention terms
  fill_u32<<<nb(N, 256), 256, 0, stream>>>((unsigned*)deg, 0u, N);
  fill_u32<<<nb(N * EHID, 256), 256, 0, stream>>>((unsigned*)loop_sum, 0u, N * EHID);
  vmat_kernel<<<nb(LAYERS * EHID * HEADS, 256), 256, 0, stream>>>(gat_edge_w, att_edge, V);
  edge_kernel<<<nb(E, 128), 128, 0, stream>>>(edge_attr, dstp, ee_w, ee_b, V,
                                              a_e, loop_sum, deg, E);
  loop_kernel<<<nb(N, 128), 128, 0, stream>>>(loop_sum, deg, V, a_e, E, (int)N);

  // node encoder: h = GELU(LN(x@enc_w + enc_b))
  gemm_wmma_f32_n128<<<(unsigned)(N / 16), 256, 0, stream>>>(x, enc_w, enc_b, h,
                                                             (int)N, NODE_DIM);
  ln_kernel<<<nb(N, 8), 256, 0, stream>>>(h, nullptr, nullptr, enc_g, enc_beta,
                                          h, (int)N, 1);

  for (int l = 0; l < LAYERS; ++l) {
    gemm_wmma_f32_n128<<<(unsigned)(N / 16), 256, 0, stream>>>(
        h, gat_lin_w + (long long)l * HIDDEN * HIDDEN, nullptr, hh, (int)N, HIDDEN);
    attn_scalar_kernel<<<nb(N * 4, 256), 256, 0, stream>>>(
        hh, att_src + l * HEADS * CDIM, att_dst + l * HEADS * CDIM, a_s, a_d, (int)N);
    fill_u32<<<nb(N * 4, 256), 256, 0, stream>>>(mEnc, ENC_NEG_INF, N * 4);
    fill_u32<<<nb(N * 4, 256), 256, 0, stream>>>((unsigned*)denom, 0u, N * 4);
    fill_u32<<<nb(N * 128, 256), 256, 0, stream>>>((unsigned*)agg, 0u, N * 128);
    long long T = (E + N) * 4;
    alpha1_kernel<<<nb(T, 256), 256, 0, stream>>>(srcp, dstp, a_s, a_d, a_e, l,
                                                  alpha, mEnc, E, (int)N);
    alpha2_kernel<<<nb(T, 256), 256, 0, stream>>>(dstp, alpha, mEnc, denom, E, (int)N);
    agg_kernel<<<nb(T, 128), 128, 0, stream>>>(srcp, dstp, alpha, denom, hh, agg,
                                               E, (int)N);
    ln_kernel<<<nb(N, 8), 256, 0, stream>>>(h, agg, gat_bias + l * HIDDEN,
                                            norm_g + l * HIDDEN, norm_bv + l * HIDDEN,
                                            h, (int)N, 0);
  }

  // graph pooling (mean + max) and MLP head
  fill_u32<<<nb(BGRAPH * 128, 256), 256, 0, stream>>>((unsigned*)mean_sum, 0u, BGRAPH * 128);
  fill_u32<<<nb(BGRAPH * 128, 256), 256, 0, stream>>>(maxEnc, ENC_NEG_INF, BGRAPH * 128);
  fill_u32<<<nb(BGRAPH, 256), 256, 0, stream>>>((unsigned*)cnt, 0u, BGRAPH);
  pool_acc_kernel<<<nb(N * 128, 256), 256, 0, stream>>>(h, batch, mean_sum, maxEnc,
                                                        cnt, (int)N);
  pool_final_kernel<<<nb(BGRAPH * 256, 256), 256, 0, stream>>>(mean_sum, maxEnc, cnt, gbuf);

  dense_kernel<<<nb(BGRAPH * 128, 256), 256, 0, stream>>>(gbuf, freq_w1, freq_b1, t1,
                                                          BGRAPH, 256, 128, 1);
  dense_kernel<<<nb(BGRAPH * 64, 256), 256, 0, stream>>>(t1, freq_w2, freq_b2, t2,
                                                         BGRAPH, 128, 64, 1);
  dense_kernel<<<nb(BGRAPH * 64, 256), 256, 0, stream>>>(t2, cls_w1, cls_b1, t3,
                                                         BGRAPH, 64, 64, 1);
  dense_kernel<<<nb(BGRAPH, 64), 64, 0, stream>>>(t3, cls_w2, cls_b2, (float*)d_out,
                                                  BGRAPH, 64, 1, 2);
}